// UniversalSegmentationDecoder_10385230922496
// MI455X (gfx1250) — compile-verified
//
#include <hip/hip_runtime.h>
#include <hip/hip_bf16.h>
#include <stdint.h>

// ---------------- problem constants (from reference) ----------------
#define BATCH   8
#define NQ      200
#define NCLS    81      // 81 classes, last one dropped
#define TOPK_N  100
#define HM      256     // mask pred H
#define WM      256     // mask pred W
#define SH      192     // scaled_h (crop)
#define SW      256     // scaled_w (crop)
#define OH      384     // origin_h
#define OW      512     // origin_w
#define MIN_SCORE 0.1f

#define SORTN   256     // bitonic sort width (NQ=200 padded to 256)
#define ROWS    32      // output rows per resize block
#define INROWS  18      // input rows per 32-output-row slab (16 + 2 halo)

typedef float v4f __attribute__((ext_vector_type(4)));
typedef float v2f __attribute__((ext_vector_type(2)));

// ---------------- CDNA5 async global->LDS helpers ----------------
__device__ __forceinline__ void async_load_lds_b128(uint32_t lds_byte_addr,
                                                    const void* gaddr) {
    // per-lane 16B copy from the global address in vaddr to the LDS address in vdst; tracked by ASYNCcnt
    asm volatile("global_load_async_to_lds_b128 %0, %1, off"
                 :: "v"(lds_byte_addr),
                    "v"((unsigned long long)(uintptr_t)gaddr)
                 : "memory");
}

__device__ __forceinline__ void wait_asynccnt0() {
#if __has_builtin(__builtin_amdgcn_s_wait_asynccnt)
    __builtin_amdgcn_s_wait_asynccnt(0);
#else
    asm volatile("s_wait_asynccnt 0" ::: "memory");
#endif
}

// ---------------- kernel 1: scores, argmax, top-k, small outputs ----------------
__global__ __launch_bounds__(SORTN)
void score_topk_kernel(const float* __restrict__ class_preds,
                       float* __restrict__ out_scores,
                       float* __restrict__ out_classes,
                       float* __restrict__ out_valid,
                       int*   __restrict__ ws_topidx,
                       float* __restrict__ ws_validf) {
    __shared__ float s_score[SORTN];
    __shared__ int   s_idx[SORTN];
    __shared__ int   s_cls[SORTN];

    const int b = blockIdx.x;
    const int t = threadIdx.x;

    float score = -1.0f;   // padding sorts below any softmax prob (>0)
    int   cls   = 0;
    if (t < NQ) {
        const float* p = class_preds + ((size_t)(b * NQ + t)) * NCLS;
        float m = -1e30f, best = -1e30f;
        int bi = 0;
        #pragma unroll 1
        for (int c = 0; c < NCLS; ++c) {
            float v = p[c];
            m = fmaxf(m, v);
            if (c < NCLS - 1 && v > best) { best = v; bi = c; }  // first-max tiebreak
        }
        float s = 0.0f;
        #pragma unroll 1
        for (int c = 0; c < NCLS; ++c) s += __expf(p[c] - m);
        score = __expf(best - m) / s;   // max softmax prob over first 80 classes
        cls = bi;
    }
    s_score[t] = score;
    s_idx[t]   = t;
    s_cls[t]   = cls;
    __syncthreads();

    // bitonic sort, descending by score, ascending index on ties (matches top_k)
    for (int k = 2; k <= SORTN; k <<= 1) {
        for (int j = k >> 1; j > 0; j >>= 1) {
            const int ixj = t ^ j;
            if (ixj > t) {
                const bool dirDesc = ((t & k) == 0);
                float sa = s_score[t],  sb = s_score[ixj];
                int   ia = s_idx[t],    ib = s_idx[ixj];
                const bool aGreater = (sa > sb) || (sa == sb && ia < ib);
                if (dirDesc ? !aGreater : aGreater) {
                    s_score[t] = sb; s_score[ixj] = sa;
                    s_idx[t]   = ib; s_idx[ixj]   = ia;
                    int ca = s_cls[t];
                    s_cls[t] = s_cls[ixj]; s_cls[ixj] = ca;
                }
            }
            __syncthreads();
        }
    }

    if (t < TOPK_N) {
        const int   o  = b * TOPK_N + t;
        const float s  = s_score[t];
        const bool  ok = s > MIN_SCORE;
        out_scores[o]  = ok ? s : 0.0f;
        out_classes[o] = ok ? (float)s_cls[t] : -1.0f;
        out_valid[o]   = ok ? 1.0f : 0.0f;
        ws_topidx[o]   = s_idx[t];
        ws_validf[o]   = ok ? 1.0f : 0.0f;
    }
}

// ---------------- kernel 2: gather + sigmoid + 2x bilinear resize ----------------
// grid.x = BATCH*TOPK_N (one (b,rank) per x), grid.y = OH/ROWS = 12
__global__ __launch_bounds__(256)
void resize_kernel(const float* __restrict__ mask_preds,
                   const int*   __restrict__ ws_topidx,
                   const float* __restrict__ ws_validf,
                   float*       __restrict__ out_masks) {
    __shared__ float tile[INROWS * SW];   // 18 KB, sigmoid-ed input slab

    const int bq = blockIdx.x;            // b*TOPK_N + rank
    const int b  = bq / TOPK_N;
    const int gy = blockIdx.y;            // output-row slab index
    const int t  = threadIdx.x;

    float* outp = out_masks + ((size_t)bq * OH + (size_t)gy * ROWS) * OW;
    const float vf = ws_validf[bq];

    if (vf == 0.0f) {
        // invalid query: zero output, skip all reads; NT streaming stores
        v4f z = (v4f)0.0f;
        for (int p = t; p < ROWS * OW / 4; p += 256)
            __builtin_nontemporal_store(z, reinterpret_cast<v4f*>(outp) + p);
        return;
    }

    const int q = ws_topidx[bq];
    const float* src = mask_preds + ((size_t)(b * NQ + q)) * (HM * WM);
    const int base = gy * 16 - 1;         // first (unclamped) input row of slab

    // --- async-stage 18 rows x 256 cols straight into LDS (b128 per lane) ---
    const uint32_t lds0 = (uint32_t)(uintptr_t)(&tile[0]);
    const int NT = INROWS * (SW / 4);     // 1152 x 16B transfers
    for (int i = t; i < NT; i += 256) {
        const int r  = i >> 6;            // row slot 0..17
        const int c4 = i & 63;            // float4 column
        int grow = base + r;
        grow = grow < 0 ? 0 : (grow > SH - 1 ? SH - 1 : grow);   // edge clamp
        const float* g = src + (size_t)grow * WM + (size_t)c4 * 4;
        async_load_lds_b128(lds0 + (uint32_t)i * 16u, g);
    }
    wait_asynccnt0();
    __syncthreads();

    // --- sigmoid once per input pixel, in place, b128 LDS ops ---
    {
        v4f* tp = reinterpret_cast<v4f*>(tile);
        for (int i = t; i < INROWS * SW / 4; i += 256) {
            v4f v = tp[i];
            v.x = 1.0f / (1.0f + __expf(-v.x));
            v.y = 1.0f / (1.0f + __expf(-v.y));
            v.z = 1.0f / (1.0f + __expf(-v.z));
            v.w = 1.0f / (1.0f + __expf(-v.w));
            tp[i] = v;
        }
    }
    __syncthreads();

    // --- half-pixel 2x bilinear, 4 output pixels per iteration ---
    // output cols [4c,4c+3] need input cols {2c-1, 2c, 2c+1, 2c+2}; y-blend shared.
    for (int p = t; p < ROWS * (OW / 4); p += 256) {
        const int oyl = p >> 7;           // /128 quads per row
        const int c   = p & 127;
        const int oy  = gy * ROWS + oyl;

        const int hy = oy >> 1;
        int   y0, y1; float wy0, wy1;
        if (oy & 1) { y0 = hy;     y1 = hy + 1; wy0 = 0.75f; wy1 = 0.25f; }
        else        { y0 = hy - 1; y1 = hy;     wy0 = 0.25f; wy1 = 0.75f; }
        const int s0 = y0 - base;         // slab slots, in [0,17] by construction
        const int s1 = y1 - base;
        const float* r0 = &tile[s0 * SW];
        const float* r1 = &tile[s1 * SW];

        int xm = 2 * c - 1; xm = xm < 0 ? 0 : xm;            // edge clamp
        int xp = 2 * c + 2; xp = xp > SW - 1 ? SW - 1 : xp;  // edge clamp

        // y-blend the 4 unique input columns (b64 for the aligned middle pair)
        const v2f m0 = *reinterpret_cast<const v2f*>(&r0[2 * c]);
        const v2f m1 = *reinterpret_cast<const v2f*>(&r1[2 * c]);
        const float av = wy0 * r0[xm] + wy1 * r1[xm];
        const float bv = wy0 * m0.x   + wy1 * m1.x;
        const float cv = wy0 * m0.y   + wy1 * m1.y;
        const float dv = wy0 * r0[xp] + wy1 * r1[xp];

        v4f o;
        o.x = (0.25f * av + 0.75f * bv) * vf;   // ox = 4c   (even)
        o.y = (0.75f * bv + 0.25f * cv) * vf;   // ox = 4c+1 (odd)
        o.z = (0.25f * bv + 0.75f * cv) * vf;   // ox = 4c+2 (even)
        o.w = (0.75f * cv + 0.25f * dv) * vf;   // ox = 4c+3 (odd)

        __builtin_nontemporal_store(
            o, reinterpret_cast<v4f*>(outp + (size_t)oyl * OW + 4 * c));
    }
}

// ---------------- launcher ----------------
extern "C" void kernel_launch(void* const* d_in, const int* in_sizes, int n_in,
                              void* d_out, int out_size, void* d_ws, size_t ws_size,
                              hipStream_t stream) {
    const float* mask_preds  = (const float*)d_in[0];
    const float* class_preds = (const float*)d_in[1];
    // d_in[2..5] are scaled_h/scaled_w/origin_h/origin_w scalars — shapes are
    // compile-time constants from the reference, hardcoded above.

    float* out          = (float*)d_out;
    float* out_masks    = out;                                          // B*K*OH*OW
    float* out_scores   = out + (size_t)BATCH * TOPK_N * OH * OW;       // 800
    float* out_classes  = out_scores + BATCH * TOPK_N;                  // 800
    float* out_valid    = out_classes + BATCH * TOPK_N;                 // 800

    int*   ws_topidx = (int*)d_ws;                                      // 800 ints
    float* ws_validf = (float*)((char*)d_ws + BATCH * TOPK_N * sizeof(int));

    score_topk_kernel<<<BATCH, SORTN, 0, stream>>>(
        class_preds, out_scores, out_classes, out_valid, ws_topidx, ws_validf);

    resize_kernel<<<dim3(BATCH * TOPK_N, OH / ROWS, 1), 256, 0, stream>>>(
        mask_preds, ws_topidx, ws_validf, out_masks);
}